// GCLSTMModel_48868137894020
// MI455X (gfx1250) — compile-verified
//
#include <hip/hip_runtime.h>
#include <hip/hip_bf16.h>

typedef __attribute__((ext_vector_type(2))) float v2f;
typedef __attribute__((ext_vector_type(8))) float v8f;

#define WMMA4(a, b, c) \
  __builtin_amdgcn_wmma_f32_16x16x4_f32(false, (a), false, (b), (short)0, (c), false, false)

__device__ __forceinline__ int imin(int a, int b) { return a < b ? a : b; }
__device__ __forceinline__ float sigm(float x) { return 1.f / (1.f + __expf(-x)); }
__device__ __forceinline__ float tanh_fast(float x) { return 1.f - 2.f / (__expf(2.f * x) + 1.f); }

// A-fragment (16x4 f32) from LDS: row clamped to 34 (garbage rows only feed unstored D rows).
// The two per-lane K values are contiguous -> single 64-bit LDS load.
__device__ __forceinline__ v2f ldA(const float* S, int lda, int mbase, int kb, int l16, int half) {
  int r = imin(mbase + l16, 34);
  return *(const v2f*)(S + r * lda + kb + half * 2);
}

__global__ __launch_bounds__(1024, 1)
void gclstm_fused(const float* __restrict__ adj,
                  const float* __restrict__ W1, const float* __restrict__ cb1,
                  const float* __restrict__ pe1, const float* __restrict__ b1,
                  const float* __restrict__ W2, const float* __restrict__ cb2,
                  const float* __restrict__ pe2, const float* __restrict__ b2,
                  const float* __restrict__ fcw, const float* __restrict__ fcb,
                  float* __restrict__ out) {
  __shared__ __align__(16) float sX1[35 * 144];   // X1 = relu(H1); cols 140..143 pad, never read
  __shared__ __align__(16) float sBig[35 * 288];  // stage1: adjacency (lda=36); stage2/3: X2 (lda=288)

  const int tid = threadIdx.x;
  const int lane = tid & 31;
  const int l16 = lane & 15;
  const int half = (lane >> 4) & 1;
  const int w = tid >> 5;  // wave id, 0..31

  // Stage 0: zero the 35x36 adjacency region (K-pad col 35 must be 0), then copy A unconditionally.
  for (int i = tid; i < 35 * 36; i += 1024) sBig[i] = 0.f;
  __syncthreads();
  for (int i = tid; i < 35 * 35; i += 1024) {
    int r = i / 35, c = i - r * 35;
    sBig[r * 36 + c] = adj[i];
  }
  __syncthreads();

  // ---------------- Stage 1: gates = A(35x35) @ W1[g](35x140), tiles 3x9 = 27 ----------------
  if (w < 27) {
    int mt = w / 9, nt = w % 9;
    int mb = mt * 16, nb = nt * 16;
    int n = imin(nb + l16, 139);          // clamped, loop-invariant
    const float* pI = W1 + 0 * 4900 + n;  // gate i
    const float* pT = W1 + 2 * 4900 + n;  // gate c (candidate)
    const float* pO = W1 + 3 * 4900 + n;  // gate o
    v8f ai = {}, at = {}, ao = {};
    for (int kc = 0; kc < 9; ++kc) {
      int kb = kc * 4;
      int k0 = imin(kb + half * 2, 34);      // k=35 row clamped; A's zero K-pad nullifies it
      int k1 = imin(kb + half * 2 + 1, 34);
      v2f a = ldA(sBig, 36, mb, kb, l16, half);
      v2f bI, bT, bO;
      bI.x = pI[k0 * 140]; bI.y = pI[k1 * 140];
      bT.x = pT[k0 * 140]; bT.y = pT[k1 * 140];
      bO.x = pO[k0 * 140]; bO.y = pO[k1 * 140];
      ai = WMMA4(a, bI, ai);
      at = WMMA4(a, bT, at);
      ao = WMMA4(a, bO, ao);
    }
    int cr = imin(nb + l16, 139);  // clamped for parameter reads
    int cs = nb + l16;             // raw store column (pad cols are real, never-read slots)
    float bbi = cb1[cr] + b1[cr];
    float bbt = cb1[280 + cr] + b1[280 + cr];
    float bbo = cb1[420 + cr] + b1[420 + cr];
    float pw = pe1[280 + cr];
#pragma unroll
    for (int v = 0; v < 8; ++v) {
      int r = mb + v + half * 8;
      float ig = sigm(ai[v] + bbi);
      float tg = tanh_fast(at[v] + bbt);
      float Cc = ig * tg;
      float og = sigm(ao[v] + bbo + pw * Cc);
      float h = fmaxf(og * tanh_fast(Cc), 0.f);
      if (r < 35) sX1[r * 144 + cs] = h;
    }
  }
  __syncthreads();

  // ---------------- Stage 2: X1(35x140) @ W2[g](140x280), tiles 3x18 = 54 ----------------
  for (int t = w; t < 54; t += 32) {
    int mt = t / 18, nt = t % 18;
    int mb = mt * 16, nb = nt * 16;
    int n = imin(nb + l16, 279);
    const float* pI = W2 + 0 * 39200 + n;
    const float* pT = W2 + 2 * 39200 + n;
    const float* pO = W2 + 3 * 39200 + n;
    v8f ai = {}, at = {}, ao = {};
    for (int kc = 0; kc < 35; ++kc) {  // K = 140 exact, no clamps
      int kb = kc * 4;
      int k = kb + half * 2;
      v2f a = ldA(sX1, 144, mb, kb, l16, half);
      v2f bI, bT, bO;
      bI.x = pI[k * 280]; bI.y = pI[(k + 1) * 280];
      bT.x = pT[k * 280]; bT.y = pT[(k + 1) * 280];
      bO.x = pO[k * 280]; bO.y = pO[(k + 1) * 280];
      ai = WMMA4(a, bI, ai);
      at = WMMA4(a, bT, at);
      ao = WMMA4(a, bO, ao);
    }
    int cr = imin(nb + l16, 279);
    int cs = nb + l16;
    float bbi = cb2[cr] + b2[cr];
    float bbt = cb2[560 + cr] + b2[560 + cr];
    float bbo = cb2[840 + cr] + b2[840 + cr];
    float pw = pe2[560 + cr];
#pragma unroll
    for (int v = 0; v < 8; ++v) {
      int r = mb + v + half * 8;
      float ig = sigm(ai[v] + bbi);
      float tg = tanh_fast(at[v] + bbt);
      float Cc = ig * tg;
      float og = sigm(ao[v] + bbo + pw * Cc);
      float h = fmaxf(og * tanh_fast(Cc), 0.f);
      if (r < 35) sBig[r * 288 + cs] = h;  // X2 (adjacency dead)
    }
  }
  __syncthreads();

  // ---------------- Stage 3: out = relu(X2(35x280) @ fcw(280x35) + fcb), tiles 3x3 = 9 ------
  if (w < 9) {
    int mt = w / 3, nt = w % 3;
    int mb = mt * 16, nb = nt * 16;
    int n = imin(nb + l16, 34);
    const float* pF = fcw + n;
    v8f acc = {};
    for (int kc = 0; kc < 70; ++kc) {  // K = 280 exact
      int kb = kc * 4;
      int k = kb + half * 2;
      v2f a = ldA(sBig, 288, mb, kb, l16, half);
      v2f b;
      b.x = pF[k * 35];
      b.y = pF[(k + 1) * 35];
      acc = WMMA4(a, b, acc);
    }
    int c = nb + l16;
    float bb = fcb[imin(c, 34)];
#pragma unroll
    for (int v = 0; v < 8; ++v) {
      int r = mb + v + half * 8;
      if (r < 35 && c < 35) out[r * 35 + c] = fmaxf(acc[v] + bb, 0.f);
    }
  }
}

extern "C" void kernel_launch(void* const* d_in, const int* in_sizes, int n_in,
                              void* d_out, int out_size, void* d_ws, size_t ws_size,
                              hipStream_t stream) {
  (void)in_sizes; (void)n_in; (void)out_size; (void)d_ws; (void)ws_size;
  const float* adj = (const float*)d_in[0];
  const float* W1  = (const float*)d_in[1];
  // d_in[2] = cheb1_W: mathematically dead (H0 = 0)
  const float* cb1 = (const float*)d_in[3];
  const float* pe1 = (const float*)d_in[4];
  const float* b1  = (const float*)d_in[5];
  const float* W2  = (const float*)d_in[6];
  // d_in[7] = cheb2_W: dead
  const float* cb2 = (const float*)d_in[8];
  const float* pe2 = (const float*)d_in[9];
  const float* b2  = (const float*)d_in[10];
  const float* fcw = (const float*)d_in[11];
  const float* fcb = (const float*)d_in[12];

  gclstm_fused<<<1, 1024, 0, stream>>>(adj, W1, cb1, pe1, b1,
                                       W2, cb2, pe2, b2, fcw, fcb, (float*)d_out);
}